// MultiHeadAttentionQuantum_65481071397690
// MI455X (gfx1250) — compile-verified
//
#include <hip/hip_runtime.h>
#include <hip/hip_bf16.h>
#include <math.h>

// out[t,e] = sum_{q<8} cos(x[t,q] + theta[q]) * W_out[e,q]
// x: [4,4096,1024] f32  theta: [8] f32  W_out: [1024,8] f32  out: [4,4096,1024] f32
//
// Store-bandwidth-bound (64 MiB out). GEMM [16384x8]x[8x1024] done with
// V_WMMA_F32_16X16X4_F32 (two K=4 steps for K=8), exact f32 accumulation.

typedef __attribute__((ext_vector_type(2))) float v2f;
typedef __attribute__((ext_vector_type(8))) float v8f;

#define NQ       8
#define E_DIM    1024
#define XSTRIDE  1024
#define MTILES   1024          // 16384 tokens / 16

// Block: 256 threads = 8 waves. Block b owns m-tile b (16 tokens).
// Wave w covers n-tiles [8w, 8w+8) -> output columns [128w, 128w+128).
__global__ __launch_bounds__(256)
void quantum_proj_wmma_kernel(const float* __restrict__ x,
                              const float* __restrict__ theta,
                              const float* __restrict__ Wout,
                              float* __restrict__ out)
{
    const int lane  = threadIdx.x & 31;
    const int wave  = threadIdx.x >> 5;
    const int mtile = blockIdx.x;                 // 0..1023
    const int m     = lane & 15;                  // token-in-tile == col-in-tile
    const int kb    = (lane >> 4) << 1;           // 0 for lanes 0-15, 2 for 16-31

    // ---- A fragment: proj tile, 16x4 layout per ISA (2 VGPRs per K-step) ----
    const int     token = mtile * 16 + m;
    const float* __restrict__ xr = x + (size_t)token * XSTRIDE;

    const float t0 = theta[kb + 0];
    const float t1 = theta[kb + 1];
    const float t4 = theta[kb + 4];
    const float t5 = theta[kb + 5];

    v2f a0, a1;                                   // K=0..3 and K=4..7 halves
    a0.x = cosf(xr[kb + 0] + t0);
    a0.y = cosf(xr[kb + 1] + t1);
    a1.x = cosf(xr[kb + 4] + t4);
    a1.y = cosf(xr[kb + 5] + t5);

    // Row base for the C-tile writeback: lanes 0-15 -> rows M=0..7,
    // lanes 16-31 -> rows M=8..15 (VGPR v holds row v / v+8).
    const int rowbase = mtile * 16 + ((lane >> 4) << 3);

#pragma unroll
    for (int nt = 0; nt < 8; ++nt) {
        const int ntile = wave * 8 + nt;          // 0..63
        const int e     = ntile * 16 + m;         // this lane's B column

        // ---- B fragment: W_out[e, kb..kb+1] / [e, kb+4..kb+5] (B = W_out^T, K x N) ----
        const float* __restrict__ wr = Wout + e * NQ;
        v2f b0, b1;
        b0.x = wr[kb + 0];
        b0.y = wr[kb + 1];
        b1.x = wr[kb + 4];
        b1.y = wr[kb + 5];

        // ---- D = A0*B0 + A1*B1 (K = 8 via two 16x16x4 f32 WMMAs) ----
        v8f c = {};
        c = __builtin_amdgcn_wmma_f32_16x16x4_f32(false, a0, false, b0,
                                                  (short)0, c, false, false);
        c = __builtin_amdgcn_wmma_f32_16x16x4_f32(false, a1, false, b1,
                                                  (short)0, c, false, false);

        // ---- Writeback: VGPR v -> out[rowbase + v, ntile*16 + (lane&15)] ----
        float* __restrict__ orow = out + (size_t)rowbase * E_DIM + ntile * 16 + m;
#pragma unroll
        for (int v = 0; v < 8; ++v) {
            orow[(size_t)v * E_DIM] = c[v];
        }
    }
}

extern "C" void kernel_launch(void* const* d_in, const int* in_sizes, int n_in,
                              void* d_out, int out_size, void* d_ws, size_t ws_size,
                              hipStream_t stream)
{
    const float* x     = (const float*)d_in[0];   // [4,4096,1024]
    const float* theta = (const float*)d_in[1];   // [8]
    const float* Wout  = (const float*)d_in[2];   // [1024,8]
    float*       out   = (float*)d_out;           // [4,4096,1024]

    quantum_proj_wmma_kernel<<<MTILES, 256, 0, stream>>>(x, theta, Wout, out);
}